// KerasMultiLIFLayerSparse_67628555043243
// MI455X (gfx1250) — compile-verified
//
#include <hip/hip_runtime.h>
#include <hip/hip_bf16.h>

typedef __attribute__((ext_vector_type(2))) float v2f;
typedef __attribute__((ext_vector_type(8))) float v8f;

#define L_LAYERS 3
#define DD 1024
#define BB 48
#define TT 200
#define S_INN 128
#define S_OUTT 128
#define KHALF 512
#define ASTR 514   // LDS row stride (floats): even (8B-aligned b64 loads), bank-spread (514%64==2)

// Pack weights for WMMA B-fragment loads:
// packW[l][kk][n] = { W[l][n][2kk], W[l][n][2kk+1] }  (float2)
// Lane with hi=lane>>4, n=lane&15 then loads B rows (k0+2hi, k0+2hi+1) for column n
// as one coalesced 8-byte load -> matches V_WMMA B layout (VGPR0: K=0/2, VGPR1: K=1/3).
__global__ __launch_bounds__(1024) void prep_weights(const float* __restrict__ W,
                                                     float2* __restrict__ packW) {
  int blk = blockIdx.x;            // l*512 + kk
  int l = blk >> 9, kk = blk & 511;
  int n = threadIdx.x;             // 0..1023
  const float* wrow = W + ((size_t)l * DD + n) * DD + 2 * kk;
  packW[(size_t)blk * DD + n] = make_float2(wrow[0], wrow[1]);
}

__global__ __launch_bounds__(1024) void copy_state(const float* __restrict__ src,
                                                   float* __restrict__ dst) {
  int i = blockIdx.x * 1024 + threadIdx.x;
  if (i < L_LAYERS * BB * DD) dst[i] = src[i];
}

// Exact reproduction of jax.lax.top_k on a binary vector:
// spiking indices ascending, then non-spiking indices ascending, truncated to 128.
__device__ void compact_rows(const float* __restrict__ src, int ct, int cl, int row0,
                             int wave, int lane,
                             float* __restrict__ out_ids, int* __restrict__ out_nums) {
  unsigned lmask = (1u << lane) - 1u;
  for (int rr = 0; rr < 2; ++rr) {
    int b = row0 + wave * 2 + rr;
    const float* row = src + (size_t)b * DD;
    float* ido = out_ids + ((size_t)(ct * L_LAYERS + cl) * BB + b) * S_OUTT;
    int cnt = 0;
    for (int i = 0; i < 32; ++i) {               // pass 1: spiking neurons, ascending
      int d = (i << 5) + lane;
      bool f = row[d] != 0.0f;
      unsigned m = __builtin_amdgcn_ballot_w32(f);
      int pos = cnt + __popc(m & lmask);
      if (f && pos < S_OUTT) ido[pos] = (float)d;
      cnt += __popc(m);
    }
    if (lane == 0)
      out_nums[(ct * L_LAYERS + cl) * BB + b] = cnt < S_OUTT ? cnt : S_OUTT;
    int c2 = cnt;
    for (int i = 0; i < 32; ++i) {               // pass 2: zero-padding indices, ascending
      if (c2 >= S_OUTT) break;                   // wave-uniform
      int d = (i << 5) + lane;
      bool f = row[d] != 0.0f;
      unsigned m = __builtin_amdgcn_ballot_w32(!f);
      int pos = c2 + __popc(m & lmask);
      if (!f && pos < S_OUTT) ido[pos] = (float)d;
      c2 += __popc(m);
    }
  }
}

// One layer of one timestep. grid = (8 n-groups, 3 row-tiles), block = 256 (8 wave32).
// Each wave computes a 16x16 output tile with V_WMMA_F32_16X16X4_F32 over K=1024
// (two K=512 LDS-staged phases), then fused LIF update; n-group-0 blocks also run
// the compaction for the previous layer's spike output.
__global__ __launch_bounds__(256) void lif_layer(
    int t, int l,
    const int* __restrict__ sp_ids, const int* __restrict__ sp_nums,   // layer-0 sparse input
    const float* __restrict__ dense_in,                                // [BB][DD] or null
    const float2* __restrict__ packW,                                  // [3*512][1024]
    float* __restrict__ state,                                         // [3][BB][DD]
    float* __restrict__ spk_out,                                       // [BB][DD]
    const float* __restrict__ decay, const float* __restrict__ thresh, // [3][DD]
    float* __restrict__ out_states, float* __restrict__ out_ids, int* __restrict__ out_nums,
    const float* __restrict__ compact_src, int compact_t, int compact_l)
{
  __shared__ float As[16 * ASTR];
  const int tid  = threadIdx.x;
  const int lane = tid & 31;
  const int wave = tid >> 5;       // 0..7
  const int ngrp = blockIdx.x;     // 0..7
  const int row0 = blockIdx.y * 16;

  const int hi = lane >> 4;                       // lane half selects K rows 0/1 vs 2/3
  const int nl = ngrp * 128 + wave * 16 + (lane & 15);

  v8f acc = {};
  for (int half = 0; half < 2; ++half) {
    __syncthreads();                              // A tile free for restage
    if (dense_in) {
      const float* src = dense_in + (size_t)row0 * DD + half * KHALF;
      for (int i = tid; i < 16 * (KHALF / 2); i += 256) {
        int m = i >> 8;                           // 256 float2 per row
        int k2 = i & 255;
        float2 v = ((const float2*)(src + (size_t)m * DD))[k2];
        As[m * ASTR + 2 * k2]     = v.x;
        As[m * ASTR + 2 * k2 + 1] = v.y;
      }
    } else {
      for (int i = tid; i < 16 * ASTR; i += 256) As[i] = 0.0f;
      __syncthreads();
      for (int s = tid; s < 16 * S_INN; s += 256) {   // scatter sparse ids (with duplicates)
        int m = s >> 7, slot = s & 127;
        int b = row0 + m;
        int num = sp_nums[t * BB + b];
        if (slot < num) {
          int id = sp_ids[((size_t)t * BB + b) * S_INN + slot];
          int kk = id - half * KHALF;
          if (kk >= 0 && kk < KHALF) atomicAdd(&As[m * ASTR + kk], 1.0f);
        }
      }
    }
    __syncthreads();

    const v2f* ap = (const v2f*)(As + (lane & 15) * ASTR + 2 * hi);
    const v2f* bp = (const v2f*)packW + (size_t)(l * 512 + half * 256 + hi) * DD + nl;
    #pragma unroll 8
    for (int kk = 0; kk < 128; ++kk) {
      v2f a = ap[0];                   // A[m][k0+2hi .. +1]   (ds_load_b64)
      v2f b = bp[0];                   // B[k0+2hi..+1][n]     (global_load_b64, coalesced)
      acc = __builtin_amdgcn_wmma_f32_16x16x4_f32(false, a, false, b,
                                                  (short)0, acc, false, false);
      ap += 2;                         // advance 4 floats in K
      bp += 2048;                      // advance 2 kk-pair rows
    }
  }

  // Fused LIF update: v = state*decay + syn; spike/reset; write all outputs.
  {
    float dc = decay[l * DD + nl];
    float th = thresh[l * DD + nl];
    float* st = state + (size_t)l * BB * DD;
    float* so = out_states + ((size_t)t * L_LAYERS + l) * BB * DD;
    #pragma unroll
    for (int r = 0; r < 8; ++r) {
      int b = row0 + hi * 8 + r;                 // C/D layout: M = r + 8*hi, N = lane&15
      size_t idx = (size_t)b * DD + nl;
      float v = st[idx] * dc + acc[r];
      bool sp = v > th;
      so[idx]      = v;                          // states_seq stores pre-reset v
      st[idx]      = sp ? 0.0f : v;              // reset-to-zero on spike
      spk_out[idx] = sp ? 1.0f : 0.0f;           // dense spikes for next layer
    }
  }

  // Fused compaction of the previous layer's spikes (3 blocks x 16 rows covers B=48).
  if (compact_src != nullptr && ngrp == 0) {
    compact_rows(compact_src, compact_t, compact_l, row0, wave, lane, out_ids, out_nums);
  }
}

__global__ __launch_bounds__(256) void compact_tail(const float* __restrict__ src, int ct, int cl,
                                                    float* __restrict__ out_ids,
                                                    int* __restrict__ out_nums) {
  compact_rows(src, ct, cl, blockIdx.x * 16, threadIdx.x >> 5, threadIdx.x & 31,
               out_ids, out_nums);
}

extern "C" void kernel_launch(void* const* d_in, const int* in_sizes, int n_in,
                              void* d_out, int out_size, void* d_ws, size_t ws_size,
                              hipStream_t stream) {
  (void)in_sizes; (void)n_in; (void)out_size; (void)ws_size;
  const float* W       = (const float*)d_in[0];   // [3][1024][1024]
  const float* init_st = (const float*)d_in[1];   // [3][48][1024]
  const int*   sp_ids  = (const int*)d_in[2];     // [200][48][128]
  const int*   sp_nums = (const int*)d_in[3];     // [200][48][1]
  const float* decay   = (const float*)d_in[4];   // [3][1024]
  const float* thresh  = (const float*)d_in[5];   // [3][1024]

  float*  ws    = (float*)d_ws;
  float2* packW = (float2*)ws;                    // 3*512*1024 float2 = 12 MB
  size_t  off   = (size_t)3 * 512 * 1024 * 2;     // in floats
  float*  state = ws + off;  off += (size_t)L_LAYERS * BB * DD;
  float*  spk   = ws + off;                       // [3][48][1024]

  float* out_f      = (float*)d_out;
  float* out_ids    = out_f;                                                   // [T][L][B][128] f32
  int*   out_nums   = (int*)(out_f + (size_t)TT * L_LAYERS * BB * S_OUTT);     // [T][L][B][1] i32
  float* out_states = out_f + (size_t)TT * L_LAYERS * BB * S_OUTT
                            + (size_t)TT * L_LAYERS * BB;                      // [T][L][B][D] f32

  prep_weights<<<dim3(L_LAYERS * 512), dim3(1024), 0, stream>>>(W, packW);
  copy_state<<<dim3((L_LAYERS * BB * DD + 1023) / 1024), dim3(1024), 0, stream>>>(init_st, state);

  const size_t SL = (size_t)BB * DD;
  dim3 grid(8, 3), blk(256);
  for (int t = 0; t < TT; ++t) {
    // layer 0: sparse scatter input; also compacts previous step's layer-2 spikes
    lif_layer<<<grid, blk, 0, stream>>>(t, 0, sp_ids, sp_nums, nullptr, packW, state,
        spk + 0 * SL, decay, thresh, out_states, out_ids, out_nums,
        (t > 0) ? (spk + 2 * SL) : nullptr, t - 1, 2);
    // layer 1: dense input = layer-0 spikes; compacts layer-0 spikes
    lif_layer<<<grid, blk, 0, stream>>>(t, 1, nullptr, nullptr, spk + 0 * SL, packW, state,
        spk + 1 * SL, decay, thresh, out_states, out_ids, out_nums,
        spk + 0 * SL, t, 0);
    // layer 2: dense input = layer-1 spikes; compacts layer-1 spikes
    lif_layer<<<grid, blk, 0, stream>>>(t, 2, nullptr, nullptr, spk + 1 * SL, packW, state,
        spk + 2 * SL, decay, thresh, out_states, out_ids, out_nums,
        spk + 1 * SL, t, 1);
  }
  // last step's layer-2 spikes
  compact_tail<<<dim3(3), blk, 0, stream>>>(spk + 2 * SL, TT - 1, 2, out_ids, out_nums);
}